// GNN_4088808865800
// MI455X (gfx1250) — compile-verified
//
#include <hip/hip_runtime.h>
#include <math.h>

// Problem constants (match reference)
#define Nn   100000
#define Ee   1600000
#define Gg   5000
#define DIN  33
#define DD   100
#define TT   12
#define EPSbn 1e-5f
#define SAP  112    // padded row pitch (>= max padded K / H), multiple of 16
#define MR   32     // node rows per block

typedef __attribute__((ext_vector_type(2))) float v2f;
typedef __attribute__((ext_vector_type(8))) float v8f;

// ---------------------------------------------------------------- utilities
__global__ void copy4_f32(const float4* __restrict__ src, float4* __restrict__ dst, int n4) {
  int i = blockIdx.x * blockDim.x + threadIdx.x;
  if (i < n4) dst[i] = src[i];
}

__global__ void zero_f32(float* __restrict__ dst, int n) {
  int i = blockIdx.x * blockDim.x + threadIdx.x;
  if (i < n) dst[i] = 0.f;
}

// Pad weight [K x H] into [112 x 112] zero-padded, K-row-pair interleaved:
//   wp[(k>>1)*(2*SAP) + 2*c + (k&1)] = w[k*H + c]
// so the WMMA B fragment (rows k,k+1 at col c, k even) is one aligned b64 load.
__global__ void pad_w(const float* __restrict__ w, float* __restrict__ wp, int K, int H) {
  int idx = blockIdx.x * blockDim.x + threadIdx.x;
  if (idx >= SAP * SAP) return;
  int kpair = idx / (2 * SAP);
  int rem = idx - kpair * (2 * SAP);
  int c = rem >> 1;
  int k = (kpair << 1) | (rem & 1);
  wp[idx] = (k < K && c < H) ? w[(size_t)k * H + c] : 0.f;
}

// ------------------------------------------------- edge scatter: agg += h[src]
// one wave per edge: coalesced row read, coalesced global_atomic_add_f32 row
// write; h (40MB) is resident in the 192MB L2 so this stays on-chip.
__global__ void edge_scatter(const float* __restrict__ h, const int* __restrict__ src,
                             const int* __restrict__ dst, float* __restrict__ agg,
                             int E, int Dm) {
  int gid = blockIdx.x * blockDim.x + threadIdx.x;
  int e = gid >> 5;
  int lane = gid & 31;
  if (e >= E) return;
  int s = src[e];
  int t = dst[e];
  const float* hr = h + (size_t)s * Dm;
  float* ar = agg + (size_t)t * Dm;
  for (int d = lane; d < Dm; d += 32)
    atomicAdd(&ar[d], hr[d]);
}

// ------------------------------------------------------------- fused GIN MLP
// block = 224 threads = 7 wave32's, owns MR=32 node rows.
//  phase0: stage agg rows (32 x K1, zero padded) into LDS; zero sZ
//  phase1: wave w -> z cols [16w,16w+16): z = sA @ w1 + b1, BN, ReLU -> sZ
//          (two 16-row accumulators share each B fragment: 2 WMMA / chunk)
//  phase2: wave w -> y cols: y = sZ @ w2 + b2, ReLU, outer BN, (ReLU)
// WMMA fp32 16x16x4; fragment layouts per ISA 7.12.2; loops have constant
// trip counts (templated shapes) and guard-free loads (padded weights).
template <int K1, int H, int OUT>
__global__ __launch_bounds__(224)
void gin_mlp(const float* __restrict__ agg, float* __restrict__ hout,
             const float* __restrict__ wp1, const float* __restrict__ b1,
             const float* __restrict__ g1, const float* __restrict__ be1,
             const float* __restrict__ m1, const float* __restrict__ v1,
             const float* __restrict__ wp2, const float* __restrict__ b2,
             const float* __restrict__ og, const float* __restrict__ ob,
             const float* __restrict__ om, const float* __restrict__ ov,
             int final_relu) {
  constexpr int KP1 = (K1 + 3) & ~3;   // padded K of GEMM1
  constexpr int KP2 = (H + 3) & ~3;    // padded K of GEMM2
  __shared__ float sA[MR * SAP];
  __shared__ float sZ[MR * SAP];
  const int tile = blockIdx.x;
  const int tid = threadIdx.x;
  const int wv = tid >> 5;             // wave 0..6
  const int lane = tid & 31;

  // phase 0: stage A tile (zero padded), zero Z tile
  for (int idx = tid; idx < MR * SAP; idx += 224) {
    int r = idx / SAP;
    int c = idx - r * SAP;
    int gr = tile * MR + r;
    sA[idx] = (gr < Nn && c < K1) ? agg[(size_t)gr * K1 + c] : 0.f;
    sZ[idx] = 0.f;
  }
  __syncthreads();

  const int m    = lane & 15;          // fragment row / col-in-tile
  const int koff = (lane >> 4) << 1;   // K sub-offset (0 | 2) per ISA layout

  // ---------------- phase 1: z = relu(bn(sA @ w1 + b1)) ----------------
  constexpr int nhid = (H + 15) >> 4;
  if (wv < nhid) {                     // wave-uniform: EXEC all ones inside
    v8f acc0 = {};
    v8f acc1 = {};
    const int col = (wv << 4) + m;
    const v2f* __restrict__ w1v = (const v2f*)wp1;
    for (int k0 = 0; k0 < KP1; k0 += 4) {
      const int kb = k0 + koff;
      v2f a0 = *(const v2f*)&sA[m * SAP + kb];
      v2f a1 = *(const v2f*)&sA[(m + 16) * SAP + kb];
      v2f b  = w1v[(kb >> 1) * SAP + col];           // rows kb,kb+1 @ col
      acc0 = __builtin_amdgcn_wmma_f32_16x16x4_f32(false, a0, false, b,
                                                   (short)0, acc0, false, false);
      acc1 = __builtin_amdgcn_wmma_f32_16x16x4_f32(false, a1, false, b,
                                                   (short)0, acc1, false, false);
    }
    if (col < H) {
      const float bb    = b1[col];
      const float scale = g1[col] * rsqrtf(v1[col] + EPSbn);
      const float mean  = m1[col];
      const float beta  = be1[col];
      #pragma unroll
      for (int i = 0; i < 8; i++) {
        const int mm = i + ((lane >> 4) << 3);
        float z0 = (acc0[i] + bb - mean) * scale + beta;
        float z1 = (acc1[i] + bb - mean) * scale + beta;
        sZ[mm * SAP + col]        = fmaxf(z0, 0.f);
        sZ[(mm + 16) * SAP + col] = fmaxf(z1, 0.f);
      }
    }
  }
  __syncthreads();

  // ------- phase 2: y = outerbn(relu(sZ @ w2 + b2)) (+ optional relu) -------
  constexpr int nout = (OUT + 15) >> 4;
  if (wv < nout) {
    v8f acc0 = {};
    v8f acc1 = {};
    const int col = (wv << 4) + m;
    const v2f* __restrict__ w2v = (const v2f*)wp2;
    for (int k0 = 0; k0 < KP2; k0 += 4) {
      const int kb = k0 + koff;
      v2f a0 = *(const v2f*)&sZ[m * SAP + kb];
      v2f a1 = *(const v2f*)&sZ[(m + 16) * SAP + kb];
      v2f b  = w2v[(kb >> 1) * SAP + col];
      acc0 = __builtin_amdgcn_wmma_f32_16x16x4_f32(false, a0, false, b,
                                                   (short)0, acc0, false, false);
      acc1 = __builtin_amdgcn_wmma_f32_16x16x4_f32(false, a1, false, b,
                                                   (short)0, acc1, false, false);
    }
    if (col < OUT) {
      const float bb    = b2[col];
      const float scale = og[col] * rsqrtf(ov[col] + EPSbn);
      const float mean  = om[col];
      const float beta  = ob[col];
      #pragma unroll
      for (int i = 0; i < 8; i++) {
        const int mm = i + ((lane >> 4) << 3);
        const int gr0 = tile * MR + mm;
        const int gr1 = gr0 + 16;
        float y0 = fmaxf(acc0[i] + bb, 0.f);
        float y1 = fmaxf(acc1[i] + bb, 0.f);
        y0 = (y0 - mean) * scale + beta;
        y1 = (y1 - mean) * scale + beta;
        if (final_relu) { y0 = fmaxf(y0, 0.f); y1 = fmaxf(y1, 0.f); }
        if (gr0 < Nn) hout[(size_t)gr0 * DD + col] = y0;
        if (gr1 < Nn) hout[(size_t)gr1 * DD + col] = y1;
      }
    }
  }
}

// ------------------------------------------ per-graph offsets (batch sorted)
__global__ void offsets_k(const int* __restrict__ batch, int* __restrict__ offs, int N, int G) {
  int i = blockIdx.x * blockDim.x + threadIdx.x;
  if (i >= N) return;
  int b = batch[i];
  if (i == 0) {
    for (int g = 0; g <= b; g++) offs[g] = 0;
  } else {
    int p = batch[i - 1];
    for (int g = p + 1; g <= b; g++) offs[g] = i;
  }
  if (i == N - 1) {
    for (int g = b + 1; g < G; g++) offs[g] = N;
  }
}

// --------------------------------------------- global add pool: wave per node
__global__ void pool_k(const float* __restrict__ h, const int* __restrict__ batch,
                       float* __restrict__ hg, int N) {
  int gid = blockIdx.x * blockDim.x + threadIdx.x;
  int node = gid >> 5;
  int lane = gid & 31;
  if (node >= N) return;
  int g = batch[node];
  const float* hr = h + (size_t)node * DD;
  float* gr = hg + (size_t)g * DD;
  for (int d = lane; d < DD; d += 32)
    atomicAdd(&gr[d], hr[d]);
}

// ---------------------------------------- select node, head linear + sigmoid
__global__ void select_head(const float* __restrict__ h, const float* __restrict__ hg,
                            const int* __restrict__ atom_num, const int* __restrict__ offs,
                            const float* __restrict__ hw, const float* __restrict__ hb,
                            float* __restrict__ out, float* __restrict__ o_sel,
                            float* __restrict__ o_w, float* __restrict__ o_cat) {
  int g = blockIdx.x;
  __shared__ float s[2 * DD];
  int sel = atom_num[g] + offs[g];
  sel = sel < 0 ? 0 : (sel > Nn - 1 ? Nn - 1 : sel);  // JAX gather clamp
  for (int d = threadIdx.x; d < DD; d += blockDim.x) {
    float hs = h[(size_t)sel * DD + d];
    float hwv = 0.5f * hg[(size_t)g * DD + d];
    s[d] = hs;
    s[DD + d] = hwv;
    o_sel[(size_t)g * DD + d] = hs;
    o_w[(size_t)g * DD + d] = hwv;
    o_cat[(size_t)g * 2 * DD + d] = hs;
    o_cat[(size_t)g * 2 * DD + DD + d] = hwv;
  }
  __syncthreads();
  for (int t = threadIdx.x; t < TT; t += blockDim.x) {
    float acc = hb[t];
    for (int d = 0; d < 2 * DD; d++) acc += s[d] * hw[d * TT + t];
    out[(size_t)g * TT + t] = 1.f / (1.f + expf(-acc));
  }
}

// --------------------------------------------------------------------- launch
extern "C" void kernel_launch(void* const* d_in, const int* in_sizes, int n_in,
                              void* d_out, int out_size, void* d_ws, size_t ws_size,
                              hipStream_t stream) {
  (void)in_sizes; (void)n_in; (void)out_size; (void)ws_size;

  const float* x        = (const float*)d_in[0];
  const int*   ei       = (const int*)d_in[1];
  const int*   batch    = (const int*)d_in[2];
  const int*   atom_num = (const int*)d_in[3];
  const float* l0_w1  = (const float*)d_in[4];
  const float* l0_b1  = (const float*)d_in[5];
  const float* l0_bng = (const float*)d_in[6];
  const float* l0_bnb = (const float*)d_in[7];
  const float* l0_bnm = (const float*)d_in[8];
  const float* l0_bnv = (const float*)d_in[9];
  const float* l0_w2  = (const float*)d_in[10];
  const float* l0_b2  = (const float*)d_in[11];
  const float* ws1  = (const float*)d_in[12];
  const float* bs1  = (const float*)d_in[13];
  const float* bng  = (const float*)d_in[14];
  const float* bnb  = (const float*)d_in[15];
  const float* bnm  = (const float*)d_in[16];
  const float* bnv  = (const float*)d_in[17];
  const float* ws2  = (const float*)d_in[18];
  const float* bs2  = (const float*)d_in[19];
  const float* obng = (const float*)d_in[20];
  const float* obnb = (const float*)d_in[21];
  const float* obnm = (const float*)d_in[22];
  const float* obnv = (const float*)d_in[23];
  const float* head_w = (const float*)d_in[24];
  const float* head_b = (const float*)d_in[25];

  const int* src = ei;        // edge_index[0]
  const int* dst = ei + Ee;   // edge_index[1]

  // workspace carve-up
  char* wsb = (char*)d_ws;
  float* bufA = (float*)wsb;                                     // N*DD (h)
  float* bufB = (float*)(wsb + (size_t)Nn * DD * 4);             // N*DD (agg)
  int*   offs = (int*)(wsb + (size_t)2 * Nn * DD * 4);           // G ints
  float* hg   = (float*)(wsb + (size_t)2 * Nn * DD * 4 + 20480); // G*DD
  float* wpad = (float*)(wsb + (size_t)2 * Nn * DD * 4 + 20480 + (size_t)Gg * DD * 4);
  const size_t WPSZ = (size_t)SAP * SAP;  // floats per padded matrix
  float* wp_l0w1 = wpad;                  // 8 padded matrices
  float* wp_l0w2 = wpad + WPSZ;
  float* wp_w1   = wpad + 2 * WPSZ;       // [3] * WPSZ
  float* wp_w2   = wpad + 5 * WPSZ;       // [3] * WPSZ

  float* out   = (float*)d_out;
  float* o_sel = out + (size_t)Gg * TT;
  float* o_w   = o_sel + (size_t)Gg * DD;
  float* o_cat = o_w + (size_t)Gg * DD;

  const long long ethreads = (long long)Ee * 32;
  const unsigned eblocks = (unsigned)((ethreads + 255) / 256);
  const unsigned mlp_blocks = (Nn + MR - 1) / MR;
  const unsigned padb = (SAP * SAP + 255) / 256;

  // pad all weights (guard-free WMMA B loads later)
  pad_w<<<padb, 256, 0, stream>>>(l0_w1, wp_l0w1, DIN, DIN);
  pad_w<<<padb, 256, 0, stream>>>(l0_w2, wp_l0w2, DIN, DD);
  for (int i = 0; i < 3; i++) {
    pad_w<<<padb, 256, 0, stream>>>(ws1 + (size_t)i * DD * DD, wp_w1 + i * WPSZ, DD, DD);
    pad_w<<<padb, 256, 0, stream>>>(ws2 + (size_t)i * DD * DD, wp_w2 + i * WPSZ, DD, DD);
  }

  // ------------- layer 0 (33 -> 33 -> 100) -------------
  copy4_f32<<<(Nn * DIN / 4 + 255) / 256, 256, 0, stream>>>((const float4*)x, (float4*)bufB, Nn * DIN / 4);
  edge_scatter<<<eblocks, 256, 0, stream>>>(x, src, dst, bufB, Ee, DIN);
  gin_mlp<DIN, DIN, DD><<<mlp_blocks, 224, 0, stream>>>(bufB, bufA,
      wp_l0w1, l0_b1, l0_bng, l0_bnb, l0_bnm, l0_bnv,
      wp_l0w2, l0_b2, obng, obnb, obnm, obnv, 1);

  // ------------- layers 1..3 (100 -> 100 -> 100) -------------
  for (int L = 1; L < 4; L++) {
    int i = L - 1;
    copy4_f32<<<(Nn * DD / 4 + 255) / 256, 256, 0, stream>>>((const float4*)bufA, (float4*)bufB, Nn * DD / 4);
    edge_scatter<<<eblocks, 256, 0, stream>>>(bufA, src, dst, bufB, Ee, DD);
    gin_mlp<DD, DD, DD><<<mlp_blocks, 224, 0, stream>>>(bufB, bufA,
        wp_w1 + i * WPSZ, bs1 + (size_t)i * DD,
        bng + (size_t)i * DD, bnb + (size_t)i * DD,
        bnm + (size_t)i * DD, bnv + (size_t)i * DD,
        wp_w2 + i * WPSZ, bs2 + (size_t)i * DD,
        obng + (size_t)L * DD, obnb + (size_t)L * DD,
        obnm + (size_t)L * DD, obnv + (size_t)L * DD,
        (L < 3) ? 1 : 0);
  }

  // ------------- pooling + head -------------
  offsets_k<<<(Nn + 255) / 256, 256, 0, stream>>>(batch, offs, Nn, Gg);
  zero_f32<<<(Gg * DD + 255) / 256, 256, 0, stream>>>(hg, Gg * DD);
  pool_k<<<((unsigned)((long long)Nn * 32 + 255) / 256), 256, 0, stream>>>(bufA, batch, hg, Nn);
  select_head<<<Gg, 128, 0, stream>>>(bufA, hg, atom_num, offs, head_w, head_b,
                                      out, o_sel, o_w, o_cat);
}